// QGATLayer_v2_88905823027441
// MI455X (gfx1250) — compile-verified
//
#include <hip/hip_runtime.h>
#include <hip/hip_bf16.h>
#include <cstddef>

#define NND   8192
#define FIN   512
#define FOUT  128
#define KB    128
#define WPAD  132   // padded LDS row stride (floats): bank = (4*r + k) % 64 -> conflict-free frags
#define NEGC  (-9.0e15f)

typedef __attribute__((ext_vector_type(2))) float v2f;
typedef __attribute__((ext_vector_type(8))) float v8f;

__device__ __forceinline__ v8f wmma_f32_4(v2f a, v2f b, v8f c) {
  // D(16x16,f32) = A(16x4,f32) x B(4x16,f32) + C
  return __builtin_amdgcn_wmma_f32_16x16x4_f32(false, a, false, b, (short)0, c, false, false);
}

__device__ __forceinline__ float masked_leaky(int a, float e) {
  float lv = e > 0.f ? e : 0.2f * e;
  return (a > 0) ? lv : NEGC;
}

// ---------------- K1: hamilton = make_quaternion_mul(W) ----------------
__global__ void k_hamilton(const float* __restrict__ W, float* __restrict__ ham) {
  int idx = blockIdx.x * blockDim.x + threadIdx.x;           // 512*128
  int col = idx & 127, row = idx >> 7;
  int br = row >> 7, rr = row & 127;
  int bc = col >> 5, cc = col & 31;
  const int   comp[4][4] = {{0,1,2,3},{1,0,3,2},{2,3,0,1},{3,2,1,0}};
  const float sgn [4][4] = {{ 1.f, 1.f, 1.f, 1.f},
                            {-1.f, 1.f, 1.f,-1.f},
                            {-1.f,-1.f, 1.f, 1.f},
                            {-1.f, 1.f,-1.f, 1.f}};
  ham[idx] = sgn[br][bc] * W[rr * 128 + comp[br][bc] * 32 + cc];
}

// ---------------- K2: h = input @ hamilton  (8192x512x128, f32 WMMA) ----------------
__global__ __launch_bounds__(128) void k_gemm1(const float* __restrict__ inp,
                                               const float* __restrict__ ham,
                                               float* __restrict__ h) {
  int wid  = blockIdx.x * 4 + (threadIdx.x >> 5);   // 4096 waves total
  int lane = threadIdx.x & 31;
  int mtile = wid >> 3;            // 0..511
  int ntile = wid & 7;             // 0..7
  int r    = lane & 15;
  int half = lane >> 4;
  int arow = mtile * 16 + r;
  int bcol = ntile * 16 + r;
  v8f acc = {};
  for (int k0 = 0; k0 < FIN; k0 += 4) {
    int k = k0 + 2 * half;
    v2f a = *(const v2f*)(inp + (size_t)arow * FIN + k);   // A[r][k..k+1]
    v2f b;
    b.x = ham[(size_t)k * FOUT + bcol];                    // B[k][n]
    b.y = ham[(size_t)(k + 1) * FOUT + bcol];
    acc = wmma_f32_4(a, b, acc);
  }
#pragma unroll
  for (int vv = 0; vv < 8; ++vv)
    h[(size_t)(mtile * 16 + vv + 8 * half) * FOUT + ntile * 16 + r] = acc[vv];
}

// ---------------- K3: s1 = h @ a1, s2 = h @ a2 ----------------
__global__ void k_sv(const float* __restrict__ h, const float* __restrict__ a,
                     float* __restrict__ s1, float* __restrict__ s2) {
  int row = blockIdx.x * blockDim.x + threadIdx.x;   // 8192 threads
  const float4* hr = (const float4*)(h + (size_t)row * FOUT);
  const float4* a1 = (const float4*)a;
  const float4* a2 = (const float4*)(a + FOUT);
  float acc1 = 0.f, acc2 = 0.f;
#pragma unroll
  for (int q = 0; q < 32; ++q) {
    float4 hv = hr[q], w1 = a1[q], w2 = a2[q];
    acc1 += hv.x * w1.x + hv.y * w1.y + hv.z * w1.z + hv.w * w1.w;
    acc2 += hv.x * w2.x + hv.y * w2.y + hv.z * w2.z + hv.w * w2.w;
  }
  s1[row] = acc1; s2[row] = acc2;
}

// ---------------- K4: flash-attention fused  h' = softmax(mask(e)) @ h ----------------
// Block: 256 threads (8 waves) owns 16 rows x 128 cols. SINGLE pass over adj.
// Online softmax: running max m[16] / denom d[16] in LDS, WMMA accumulators rescaled
// by exp(m_old - m_new) per 128-j block.
__global__ __launch_bounds__(256) void k_attn(const int* __restrict__ adj,
                                              const float* __restrict__ h,
                                              const float* __restrict__ s1,
                                              const float* __restrict__ s2,
                                              float* __restrict__ hprime) {
  __shared__ float wtile[16 * WPAD];          // exp-weights, A-matrix tiles
  __shared__ float htileT[KB * WPAD];         // h block, transposed: htileT[c][j]
  __shared__ float s1s[16], ms[16], dsum[16], scl[16];

  int i0 = blockIdx.x * 16;
  int t  = threadIdx.x;
  if (t < 16) {
    s1s[t]  = s1[i0 + t];
    ms[t]   = NEGC;     // matches the masked value -> all-masked rows go uniform
    dsum[t] = 0.f;
    scl[t]  = 1.f;
  }
  __syncthreads();

  int lane = t & 31;
  int wave = t >> 5;            // 0..7 -> column tile (16 cols each)
  int r    = lane & 15;
  int half = lane >> 4;

  // weight-phase mapping: 16 threads per row, 8 contiguous j's each
  int row   = t >> 4;           // 0..15  (lanes 0-15 of wave w -> row 2w, 16-31 -> 2w+1)
  int chunk = t & 15;           // 0..15
  int jj0   = chunk * 8;
  const size_t arowbase = (size_t)(i0 + row) * NND;

  // h-staging mapping: 2 threads per j row, 64 cols each
  int jloc = t >> 1;
  int c0   = (t & 1) * 64;

  v8f acc = {};

  for (int jb = 0; jb < NND; jb += KB) {
    // ---- stage h[jb..jb+127][:] transposed into LDS ----
    const float4* src = (const float4*)(h + (size_t)(jb + jloc) * FOUT + c0);
#pragma unroll
    for (int q = 0; q < 16; ++q) {
      float4 hv = src[q];
      int c = c0 + q * 4;
      htileT[(c + 0) * WPAD + jloc] = hv.x;
      htileT[(c + 1) * WPAD + jloc] = hv.y;
      htileT[(c + 2) * WPAD + jloc] = hv.z;
      htileT[(c + 3) * WPAD + jloc] = hv.w;
    }

    // ---- online-softmax weight computation (single adj pass) ----
    const int4*   ap = (const int4*)(adj + arowbase + jb + jj0);
    const float4* sp = (const float4*)(s2 + jb + jj0);
    int4   a0 = ap[0], a1 = ap[1];
    float4 sa = sp[0], sb = sp[1];
    if (jb + KB < NND)   // prefetch next adj segment (global_prefetch_b8)
      __builtin_prefetch((const char*)(adj + arowbase + jb + KB + jj0), 0, 1);

    float s1v = s1s[row];
    float val[8];
    val[0] = masked_leaky(a0.x, s1v + sa.x);
    val[1] = masked_leaky(a0.y, s1v + sa.y);
    val[2] = masked_leaky(a0.z, s1v + sa.z);
    val[3] = masked_leaky(a0.w, s1v + sa.w);
    val[4] = masked_leaky(a1.x, s1v + sb.x);
    val[5] = masked_leaky(a1.y, s1v + sb.y);
    val[6] = masked_leaky(a1.z, s1v + sb.z);
    val[7] = masked_leaky(a1.w, s1v + sb.w);

    float bmax = val[0];
#pragma unroll
    for (int q = 1; q < 8; ++q) bmax = fmaxf(bmax, val[q]);
#pragma unroll
    for (int msk = 8; msk >= 1; msk >>= 1)
      bmax = fmaxf(bmax, __shfl_xor(bmax, msk, 16));   // butterfly: all 16 lanes get row max

    float m_old = ms[row];
    float m_new = fmaxf(m_old, bmax);

    float w[8], lsum = 0.f;
#pragma unroll
    for (int q = 0; q < 8; ++q) { w[q] = __expf(val[q] - m_new); lsum += w[q]; }
    float bsum = lsum;
#pragma unroll
    for (int msk = 8; msk >= 1; msk >>= 1)
      bsum += __shfl_xor(bsum, msk, 16);

    // store weights (two b128 LDS stores)
    float4* wdst = (float4*)&wtile[row * WPAD + jj0];
    wdst[0] = make_float4(w[0], w[1], w[2], w[3]);
    wdst[1] = make_float4(w[4], w[5], w[6], w[7]);

    if (chunk == 0) {
      float sc = __expf(m_old - m_new);     // 0 when NEG prefix gets superseded
      scl[row]  = sc;
      dsum[row] = dsum[row] * sc + bsum;
      ms[row]   = m_new;
    }
    __syncthreads();

    // ---- rescale accumulators, then 32 WMMA steps ----
#pragma unroll
    for (int vv = 0; vv < 8; ++vv) acc[vv] *= scl[vv + 8 * half];
#pragma unroll
    for (int kk = 0; kk < KB; kk += 4) {
      v2f a = *(const v2f*)&wtile[r * WPAD + kk + 2 * half];                 // A[r][k..k+1]
      v2f b = *(const v2f*)&htileT[(wave * 16 + r) * WPAD + kk + 2 * half];  // B[k..k+1][n]
      acc = wmma_f32_4(a, b, acc);
    }
    __syncthreads();
  }

#pragma unroll
  for (int vv = 0; vv < 8; ++vv) {
    int ro = vv + 8 * half;
    hprime[(size_t)(i0 + ro) * FOUT + wave * 16 + r] = acc[vv] / dsum[ro];
  }
}

// ---------------- K5: per-block partial column sums / sumsq (deterministic) --------
__global__ __launch_bounds__(256) void k_bnpart(const float* __restrict__ hp,
                                                float* __restrict__ partial) {
  __shared__ float ssum[256], ssq[256];
  int b = blockIdx.x, t = threadIdx.x;
  int c = t & 127, rh = t >> 7;
  float sum = 0.f, sq = 0.f;
  for (int q = 0; q < 32; ++q) {
    int row = b * 64 + rh * 32 + q;
    float v = hp[(size_t)row * FOUT + c];
    sum += v; sq += v * v;
  }
  ssum[t] = sum; ssq[t] = sq;
  __syncthreads();
  if (t < 128) {
    partial[(size_t)b * FOUT + c]         = ssum[t] + ssum[t + 128];
    partial[(size_t)(128 + b) * FOUT + c] = ssq[t]  + ssq[t + 128];
  }
}

// ---------------- K6: finalize mean / inv-std ----------------
__global__ void k_bnstats(const float* __restrict__ partial,
                          float* __restrict__ meanv, float* __restrict__ invstd) {
  int c = threadIdx.x;   // 128 threads, 1 block
  float s = 0.f, q = 0.f;
  for (int b = 0; b < 128; ++b) {
    s += partial[(size_t)b * FOUT + c];
    q += partial[(size_t)(128 + b) * FOUT + c];
  }
  float mean = s * (1.0f / NND);
  float var  = q * (1.0f / NND) - mean * mean;   // population var, matches jnp.var
  meanv[c]  = mean;
  invstd[c] = rsqrtf(var + 1e-5f);
}

// ---------------- K7: BN affine + ELU ----------------
__global__ void k_bnapply(const float* __restrict__ hp,
                          const float* __restrict__ meanv, const float* __restrict__ invstd,
                          const float* __restrict__ gamma, const float* __restrict__ beta,
                          float* __restrict__ out) {
  int idx = blockIdx.x * blockDim.x + threadIdx.x;   // 8192*128
  int c = idx & 127;
  float x = (hp[idx] - meanv[c]) * invstd[c] * gamma[c] + beta[c];
  out[idx] = x > 0.f ? x : (__expf(x) - 1.0f);       // ELU(alpha=1)
}

extern "C" void kernel_launch(void* const* d_in, const int* in_sizes, int n_in,
                              void* d_out, int out_size, void* d_ws, size_t ws_size,
                              hipStream_t stream) {
  const float* inp   = (const float*)d_in[0];   // 8192x512
  const int*   adj   = (const int*)  d_in[1];   // 8192x8192
  const float* W     = (const float*)d_in[2];   // 128x128
  const float* a     = (const float*)d_in[3];   // 256x1
  const float* gamma = (const float*)d_in[4];   // 128
  const float* beta  = (const float*)d_in[5];   // 128

  float* ws      = (float*)d_ws;
  float* ham     = ws;                    // 512*128      = 65536
  float* h       = ham + 65536;           // 8192*128     = 1048576
  float* s1      = h + 1048576;           // 8192
  float* s2      = s1 + 8192;             // 8192
  float* hp      = s2 + 8192;             // 8192*128     = 1048576
  float* partial = hp + 1048576;          // 2*128*128    = 32768
  float* meanv   = partial + 32768;       // 128
  float* invstd  = meanv + 128;           // 128  (total ~8.4 MB)

  k_hamilton<<<256, 256, 0, stream>>>(W, ham);
  k_gemm1 <<<1024, 128, 0, stream>>>(inp, ham, h);
  k_sv    <<<32, 256, 0, stream>>>(h, a, s1, s2);
  k_attn  <<<NND / 16, 256, 0, stream>>>(adj, h, s1, s2, hp);
  k_bnpart<<<128, 256, 0, stream>>>(hp, partial);
  k_bnstats<<<1, 128, 0, stream>>>(partial, meanv, invstd);
  k_bnapply<<<4096, 256, 0, stream>>>(hp, meanv, invstd, gamma, beta, (float*)d_out);
}